// Nms_9165460209921
// MI455X (gfx1250) — compile-verified
//
#include <hip/hip_runtime.h>
#include <stdint.h>

#ifndef __has_builtin
#define __has_builtin(x) 0
#endif

#define BATCH   16
#define NBOX    200000
#define MAX_DET 300
#define IOU_THR 0.5f
#define BINS    4096
#define KCAND   4096

// CDNA5 async global->LDS path (device pass only; fallback keeps host pass clean)
#if defined(__HIP_DEVICE_COMPILE__) && __has_builtin(__builtin_amdgcn_global_load_async_to_lds_b64)
#define HAVE_ASYNC_LDS 1
#else
#define HAVE_ASYNC_LDS 0
#endif

#if HAVE_ASYNC_LDS
// Builtin signature (from clang diagnostic): (v2i addrspace(1)*, v2i addrspace(3)*, imm int, imm int)
typedef int v2i __attribute__((ext_vector_type(2)));
typedef __attribute__((address_space(1))) v2i gv2i_t;
typedef __attribute__((address_space(3))) v2i lv2i_t;
#endif

// ---------------- Pass 0: zero histograms ----------------
__global__ void zero_hist_kernel(int* __restrict__ hist) {
  int total = BATCH * BINS;
  for (int i = blockIdx.x * blockDim.x + threadIdx.x; i < total;
       i += gridDim.x * blockDim.x)
    hist[i] = 0;
}

// ---------------- Pass 1: per-batch 4096-bin histogram of score bits[29:18] ----
__global__ void hist_kernel(const float* __restrict__ scores, int* __restrict__ hist) {
  __shared__ int h[BINS];
  const int b = blockIdx.y;
  for (int i = threadIdx.x; i < BINS; i += blockDim.x) h[i] = 0;
  __syncthreads();
  const float* s = scores + (size_t)b * NBOX;
  const int stride = gridDim.x * blockDim.x;
  for (int i = blockIdx.x * blockDim.x + threadIdx.x; i < NBOX; i += stride) {
    __builtin_prefetch(s + i + stride, 0, 0);   // global_prefetch_b8
    unsigned key = (__float_as_uint(s[i]) >> 18) & (BINS - 1);
    atomicAdd(&h[key], 1);
  }
  __syncthreads();
  int* gh = hist + (size_t)b * BINS;
  for (int i = threadIdx.x; i < BINS; i += blockDim.x) {
    int v = h[i];
    if (v) atomicAdd(&gh[i], v);
  }
}

// ---------------- Pass 2: find cutoff bin (take all bins > cut; count < KCAND) --
__global__ void cutoff_kernel(const int* __restrict__ hist, int* __restrict__ meta) {
  int b = blockIdx.x * blockDim.x + threadIdx.x;
  if (b >= BATCH) return;
  const int* h = hist + (size_t)b * BINS;
  int acc = 0, cut = -1;
  for (int bin = BINS - 1; bin >= 0; --bin) {
    int c = h[bin];
    if (acc + c >= KCAND) { cut = bin; break; }
    acc += c;
  }
  meta[b * 8 + 0] = cut;   // take keys strictly greater than cut (cut==-1 -> all)
  meta[b * 8 + 1] = acc;   // ncand (deterministic: whole bins only)
  meta[b * 8 + 2] = 0;     // compaction counter
}

// ---------------- Pass 3: compact candidates as packed u64 keys ----------------
__global__ void compact_kernel(const float* __restrict__ scores,
                               int* __restrict__ meta,
                               unsigned long long* __restrict__ cand) {
  const int b = blockIdx.y;
  const int cut = meta[b * 8 + 0];
  const float* s = scores + (size_t)b * NBOX;
  unsigned long long* cb = cand + (size_t)b * KCAND;
  const int stride = gridDim.x * blockDim.x;
  for (int i = blockIdx.x * blockDim.x + threadIdx.x; i < NBOX; i += stride) {
    __builtin_prefetch(s + i + stride, 0, 0);
    float sc = s[i];
    unsigned key = (__float_as_uint(sc) >> 18) & (BINS - 1);
    if ((int)key > cut) {
      int pos = atomicAdd(&meta[b * 8 + 2], 1);
      if (pos < KCAND)
        cb[pos] = ((unsigned long long)__float_as_uint(sc) << 32) |
                  (unsigned long long)(0xFFFFFFFFu - (unsigned)i);
    }
  }
}

// ---------------- Pass 4: async LDS load + bitonic sort + greedy NMS -----------
__global__ __launch_bounds__(1024)
void nms_kernel(const float* __restrict__ boxes, const int* __restrict__ classes,
                const unsigned long long* __restrict__ cand,
                const int* __restrict__ meta,
                long long* __restrict__ o_sel, float* __restrict__ o_sc,
                float* __restrict__ o_bx, int* __restrict__ o_cl,
                int* __restrict__ o_nd) {
  __shared__ unsigned long long s_key[KCAND];
  __shared__ float sel_x1[MAX_DET], sel_y1[MAX_DET];
  __shared__ float sel_x2[MAX_DET], sel_y2[MAX_DET], sel_a[MAX_DET];
  __shared__ float4 s_cbox;
  __shared__ int s_nsel;
  __shared__ int s_flag;

  const int b = blockIdx.x;
  const int tid = threadIdx.x;
  int ncand = meta[b * 8 + 1];
  if (ncand > KCAND) ncand = KCAND;
  const unsigned long long* gc = cand + (size_t)b * KCAND;

  // Load candidate keys into LDS via CDNA5 async global->LDS DMA when available.
  for (int i = tid; i < KCAND; i += blockDim.x) {
    if (i < ncand) {
#if HAVE_ASYNC_LDS
      __builtin_amdgcn_global_load_async_to_lds_b64(
          (gv2i_t*)(gc + i), (lv2i_t*)&s_key[i], 0, 0);
#else
      s_key[i] = gc[i];
#endif
    } else {
      s_key[i] = 0ull;
    }
  }
#if HAVE_ASYNC_LDS
#if __has_builtin(__builtin_amdgcn_s_wait_asynccnt)
  __builtin_amdgcn_s_wait_asynccnt(0);
#else
  asm volatile("s_wait_asynccnt 0" ::: "memory");
#endif
#endif
  if (tid == 0) s_nsel = 0;
  __syncthreads();

  // Bitonic sort, descending on packed (score_bits, ~idx) => score desc, idx asc.
  for (unsigned k = 2; k <= KCAND; k <<= 1) {
    for (unsigned j = k >> 1; j > 0; j >>= 1) {
      for (unsigned i = tid; i < KCAND; i += blockDim.x) {
        unsigned p = i ^ j;
        if (p > i) {
          unsigned long long a = s_key[i], c = s_key[p];
          bool up = ((i & k) == 0);
          if (up ? (a < c) : (a > c)) { s_key[i] = c; s_key[p] = a; }
        }
      }
      __syncthreads();
    }
  }

  const float4* bx4 = (const float4*)boxes + (size_t)b * NBOX;

  // Sorted greedy NMS: candidate vs selected set (<=300) in parallel across lanes.
  for (int c = 0; c < ncand; ++c) {
    int nseln = s_nsel;                 // uniform: guarded by trailing barrier
    if (nseln >= MAX_DET) break;
    unsigned long long key = s_key[c];
    int jj = (int)(0xFFFFFFFFu - (unsigned)(key & 0xFFFFFFFFull));
    if (tid == 0) { s_cbox = bx4[jj]; s_flag = 0; }
    __syncthreads();
    float4 cb = s_cbox;
    float ca = (cb.z - cb.x) * (cb.w - cb.y);
    int sup = 0;
    for (int t = tid; t < nseln; t += blockDim.x) {
      float lx = fmaxf(cb.x, sel_x1[t]);
      float ly = fmaxf(cb.y, sel_y1[t]);
      float rx = fminf(cb.z, sel_x2[t]);
      float ry = fminf(cb.w, sel_y2[t]);
      float inter = fmaxf(rx - lx, 0.f) * fmaxf(ry - ly, 0.f);
      float iou = inter / (ca + sel_a[t] - inter);
      if (iou > IOU_THR) sup = 1;
    }
    if (sup) s_flag = 1;                // benign same-value race
    __syncthreads();
    if (s_flag == 0 && tid == 0) {
      int pos = nseln;
      sel_x1[pos] = cb.x; sel_y1[pos] = cb.y;
      sel_x2[pos] = cb.z; sel_y2[pos] = cb.w;
      sel_a[pos]  = ca;
      size_t o = (size_t)b * MAX_DET + pos;
      o_sel[o] = (long long)jj;
      o_sc[o]  = __uint_as_float((unsigned)(key >> 32));
      o_bx[o * 4 + 0] = cb.x; o_bx[o * 4 + 1] = cb.y;
      o_bx[o * 4 + 2] = cb.z; o_bx[o * 4 + 3] = cb.w;
      o_cl[o] = classes[(size_t)b * NBOX + jj];
      s_nsel = pos + 1;
    }
    __syncthreads();
  }
  __syncthreads();
  int nsel = s_nsel;
  for (int m = nsel + tid; m < MAX_DET; m += blockDim.x) {
    size_t o = (size_t)b * MAX_DET + m;
    o_sel[o] = -1ll;
    o_sc[o] = 0.f;
    o_bx[o * 4 + 0] = 0.f; o_bx[o * 4 + 1] = 0.f;
    o_bx[o * 4 + 2] = 0.f; o_bx[o * 4 + 3] = 0.f;
    o_cl[o] = 2147483647;
  }
  if (tid == 0) o_nd[b] = nsel;
}

// ---------------- Host launcher ----------------
extern "C" void kernel_launch(void* const* d_in, const int* in_sizes, int n_in,
                              void* d_out, int out_size, void* d_ws, size_t ws_size,
                              hipStream_t stream) {
  const float* scores  = (const float*)d_in[0];
  const float* boxes   = (const float*)d_in[1];
  const int*   classes = (const int*)d_in[2];

  // Workspace layout: hist[BATCH*BINS] i32 | meta[BATCH*8] i32 | cand[BATCH*KCAND] u64
  char* ws = (char*)d_ws;
  int* hist = (int*)ws;
  int* meta = (int*)(ws + (size_t)BATCH * BINS * 4);
  unsigned long long* cand =
      (unsigned long long*)(ws + (size_t)BATCH * BINS * 4 + (size_t)BATCH * 8 * 4);

  // Output layout (flat, return order): i64 sel | f32 scores | f32 boxes | i32 classes | i32 n_det
  char* o = (char*)d_out;
  long long* o_sel = (long long*)o;
  float* o_sc = (float*)(o + (size_t)BATCH * MAX_DET * 8);
  float* o_bx = (float*)(o + (size_t)BATCH * MAX_DET * 12);
  int*   o_cl = (int*)(o + (size_t)BATCH * MAX_DET * 28);
  int*   o_nd = (int*)(o + (size_t)BATCH * MAX_DET * 32);

  zero_hist_kernel<<<64, 256, 0, stream>>>(hist);
  hist_kernel<<<dim3(64, BATCH), 256, 0, stream>>>(scores, hist);
  cutoff_kernel<<<1, 64, 0, stream>>>(hist, meta);
  compact_kernel<<<dim3(64, BATCH), 256, 0, stream>>>(scores, meta, cand);
  nms_kernel<<<BATCH, 1024, 0, stream>>>(boxes, classes, cand, meta,
                                         o_sel, o_sc, o_bx, o_cl, o_nd);
}